// SchNetInteraction_28587302322448
// MI455X (gfx1250) — compile-verified
//
#include <hip/hip_runtime.h>
#include <stdint.h>

// SchNet interaction block for MI455X (gfx1250), wave32, f32 WMMA 16x16x4.
// Memory-bound (~250MB HBM vs ~80 GFLOP) -> keep f32 precision.
// Gathered x-rows are staged to LDS with CDNA5 async global->LDS DMA
// (ASYNCcnt) overlapped with the first two WMMA GEMMs.

#define HIDDEN  128
#define FILTERS 128
#define GAUSS   50

typedef float v2f __attribute__((ext_vector_type(2)));
typedef float v8f __attribute__((ext_vector_type(8)));
typedef int   v4i __attribute__((vector_size(16)));  // b128 payload type

#define AS1 __attribute__((address_space(1)))
#define AS3 __attribute__((address_space(3)))

#if defined(__has_builtin)
#if __has_builtin(__builtin_amdgcn_global_load_async_to_lds_b128)
#define HAVE_ASYNC_LDS 1
#endif
#endif
#ifndef HAVE_ASYNC_LDS
#define HAVE_ASYNC_LDS 0
#endif

__device__ __forceinline__ v8f wmma_f32(v2f a, v2f b, v8f c) {
  // (neg_a, A, neg_b, B, c_mod, C, reuse_a, reuse_b)
  return __builtin_amdgcn_wmma_f32_16x16x4_f32(false, a, false, b, (short)0, c,
                                               false, false);
}

__device__ __forceinline__ float silu_f(float v) {
  return v / (1.0f + __expf(-v));
}

// Four N-tiles (64 columns) against one shared A fragment.
// B is row-major [K x FILTERS]; k0 is the per-lane K offset (kc*4 + {0,2}).
__device__ __forceinline__ void wmma_bquad(v2f a, const float* __restrict__ B,
                                           int k0, int ncol, v8f& c0, v8f& c1,
                                           v8f& c2, v8f& c3) {
  const float* r0 = B + k0 * FILTERS + ncol;
  const float* r1 = r0 + FILTERS;
  v2f b;
  b.x = r0[0];  b.y = r1[0];  c0 = wmma_f32(a, b, c0);
  b.x = r0[16]; b.y = r1[16]; c1 = wmma_f32(a, b, c1);
  b.x = r0[32]; b.y = r1[32]; c2 = wmma_f32(a, b, c2);
  b.x = r0[48]; b.y = r1[48]; c3 = wmma_f32(a, b, c3);
}

// ---------------------------------------------------------------------------
__global__ void schnet_zero_kernel(float4* __restrict__ p, long long n4) {
  long long i = (long long)blockIdx.x * blockDim.x + threadIdx.x;
  long long stride = (long long)gridDim.x * blockDim.x;
  float4 z = make_float4(0.f, 0.f, 0.f, 0.f);
  for (; i < n4; i += stride) p[i] = z;
}

// ---------------------------------------------------------------------------
// Edge kernel: 64 edges per block, 8 waves.
//   wave w -> row tile mt = w>>1 (16 edges), col half nb = (w&1)*64.
//   async:  x[src] rows -> LDS (overlapped with GEMM1/GEMM2)
//   GEMM1:  h  = silu(rbf @ w1 + b1)    (K = 50, 12 full chunks + masked tail)
//   GEMM2:  W  = (h @ w2 + b2) * cutoff (A from LDS)
//   GEMM3:  xj = xg @ l1w + l1b         (A from async-staged LDS)
//   msg = xj * W -> atomicAdd into agg[dst] (L2-resident)
// ---------------------------------------------------------------------------
__global__ __launch_bounds__(256)
void schnet_edge_kernel(const float* __restrict__ x,
                        const int*   __restrict__ edge_index,  // [2,E] int32
                        const float* __restrict__ rbf,         // [E,GAUSS]
                        const float* __restrict__ cutoff_val,  // [E]
                        const float* __restrict__ w1,
                        const float* __restrict__ b1,
                        const float* __restrict__ w2,
                        const float* __restrict__ b2,
                        const float* __restrict__ l1w,
                        const float* __restrict__ l1b,
                        float* __restrict__ agg, int E) {
  __shared__ float lds_h[64][FILTERS];   // 32 KB
  __shared__ float lds_xg[64][HIDDEN];   // 32 KB (async-staged gather)
  __shared__ float lds_cut[64];
  __shared__ int   lds_dst[64];
  __shared__ int   lds_src[64];

  const int tid  = threadIdx.x;
  const int lane = tid & 31;
  const int w    = tid >> 5;
  const int mt   = w >> 1;            // 0..3 : 16-edge row tile
  const int nb   = (w & 1) * 64;      // 0 / 64 : column half
  const int nlo  = lane & 15;
  const int khi  = (lane >> 4) << 1;  // K offset +0 / +2
  const int chi  = (lane >> 4) << 3;  // C rows +0 / +8

  const long long e0 = (long long)blockIdx.x * 64;

  if (tid < 64) {
    long long e = e0 + tid;
    if (e >= E) e = E - 1;
    lds_dst[tid] = edge_index[e];
    lds_src[tid] = edge_index[(long long)E + e];
    lds_cut[tid] = cutoff_val[e];
    long long en = e0 + 64 + tid;  // prefetch next tile's rbf rows
    if (en < (long long)E) __builtin_prefetch(rbf + en * GAUSS, 0, 1);
  }
  __syncthreads();

  // ---- stage gathered x rows into LDS (async DMA, overlapped) ----
  {
    int r  = tid >> 2;            // 0..63 : row, 4 threads per row
    int c0 = (tid & 3) * 32;      // 32 floats (= 8 x b128) per thread
    const float* src = x + (long long)lds_src[r] * HIDDEN + c0;
    float* dst = &lds_xg[r][c0];
#if HAVE_ASYNC_LDS
#pragma unroll
    for (int i = 0; i < 8; ++i) {
      __builtin_amdgcn_global_load_async_to_lds_b128(
          (AS1 v4i*)(uintptr_t)(src + i * 4),
          (AS3 v4i*)(unsigned)(uintptr_t)(dst + i * 4), 0, 0);
    }
#else
#pragma unroll
    for (int i = 0; i < 8; ++i) {
      *(float4*)(dst + i * 4) = *(const float4*)(src + i * 4);
    }
#endif
  }

  const int amr = mt * 16 + nlo;  // A-fragment row within 64-edge tile

  // ---- GEMM1: h = silu(rbf @ w1 + b1) ----
  v8f h0 = {}, h1 = {}, h2 = {}, h3 = {};
  {
    long long eA = e0 + amr;
    if (eA >= E) eA = E - 1;
    const float* arow = rbf + eA * GAUSS;
    for (int kc = 0; kc < 12; ++kc) {  // k <= 47: fully in-bounds
      int k0 = kc * 4 + khi;
      v2f a;
      a.x = arow[k0];
      a.y = arow[k0 + 1];
      wmma_bquad(a, w1, k0, nb + nlo, h0, h1, h2, h3);
    }
    {  // tail chunk: K = 48..51. Zeroed A columns kill the fake B rows.
      float mA = (lane < 16) ? 1.0f : 0.0f;
      v2f a;
      a.x = arow[48] * mA;
      a.y = arow[49] * mA;
      const float* r0 = w1 + 48 * FILTERS + nb + nlo;
      const float* r1 = r0 + FILTERS;
      v2f b;
      b.x = r0[0];  b.y = r1[0];  h0 = wmma_f32(a, b, h0);
      b.x = r0[16]; b.y = r1[16]; h1 = wmma_f32(a, b, h1);
      b.x = r0[32]; b.y = r1[32]; h2 = wmma_f32(a, b, h2);
      b.x = r0[48]; b.y = r1[48]; h3 = wmma_f32(a, b, h3);
    }
  }
#pragma unroll
  for (int t = 0; t < 4; ++t) {
    int n = nb + t * 16 + nlo;
    float bias = b1[n];
    v8f hv = (t == 0) ? h0 : (t == 1) ? h1 : (t == 2) ? h2 : h3;
#pragma unroll
    for (int r = 0; r < 8; ++r) {
      lds_h[mt * 16 + chi + r][n] = silu_f(hv[r] + bias);
    }
  }
  __syncthreads();

  // ---- GEMM2: W = h @ w2 (A from LDS) ----
  v8f w0 = {}, w1a = {}, w2a = {}, w3a = {};
  for (int kc = 0; kc < 32; ++kc) {
    int k0 = kc * 4 + khi;
    v2f a;
    a.x = lds_h[amr][k0];
    a.y = lds_h[amr][k0 + 1];
    wmma_bquad(a, w2, k0, nb + nlo, w0, w1a, w2a, w3a);
  }

  // ---- drain async gather, make LDS xg visible across waves ----
#if HAVE_ASYNC_LDS
#if __has_builtin(__builtin_amdgcn_s_wait_asynccnt)
  __builtin_amdgcn_s_wait_asynccnt(0);
#else
  asm volatile("s_wait_asynccnt 0" ::: "memory");
#endif
#endif
  __syncthreads();

  // ---- GEMM3: xj = xg @ l1w (A from LDS) ----
  v8f x0 = {}, x1 = {}, x2 = {}, x3 = {};
  for (int kc = 0; kc < 32; ++kc) {
    int k0 = kc * 4 + khi;
    v2f a;
    a.x = lds_xg[amr][k0];
    a.y = lds_xg[amr][k0 + 1];
    wmma_bquad(a, l1w, k0, nb + nlo, x0, x1, x2, x3);
  }

  // ---- msg = (xj + l1b) * (W + b2) * cutoff ; scatter-add ----
#pragma unroll
  for (int t = 0; t < 4; ++t) {
    int n = nb + t * 16 + nlo;
    float b2v = b2[n];
    float l1bv = l1b[n];
    v8f Wv = (t == 0) ? w0 : (t == 1) ? w1a : (t == 2) ? w2a : w3a;
    v8f Xv = (t == 0) ? x0 : (t == 1) ? x1 : (t == 2) ? x2 : x3;
#pragma unroll
    for (int r = 0; r < 8; ++r) {
      int m = mt * 16 + chi + r;
      float valid = ((e0 + m) < (long long)E) ? 1.0f : 0.0f;
      float Wm = (Wv[r] + b2v) * lds_cut[m];
      float xj = Xv[r] + l1bv;
      atomicAdd(&agg[(long long)lds_dst[m] * FILTERS + n], xj * Wm * valid);
    }
  }
}

// ---------------------------------------------------------------------------
// Node kernel: out = silu(agg @ l2w + l2b). 16 nodes x 128 cols per block.
// ---------------------------------------------------------------------------
__global__ __launch_bounds__(256)
void schnet_node_kernel(const float* __restrict__ agg,
                        const float* __restrict__ l2w,
                        const float* __restrict__ l2b,
                        float* __restrict__ out, int N) {
  const int tid  = threadIdx.x;
  const int lane = tid & 31;
  const int w    = tid >> 5;
  const int n0   = blockIdx.x * 16;
  const int col  = w * 16 + (lane & 15);
  const int khi  = (lane >> 4) << 1;
  const int chi  = (lane >> 4) << 3;

  int am = n0 + (lane & 15);
  if (am >= N) am = N - 1;
  const float* arow = agg + (long long)am * FILTERS;

  v8f acc = {};
  for (int kc = 0; kc < 32; ++kc) {
    int k0 = kc * 4 + khi;
    v2f a;
    a.x = arow[k0];
    a.y = arow[k0 + 1];
    v2f b;
    b.x = l2w[k0 * HIDDEN + col];
    b.y = l2w[(k0 + 1) * HIDDEN + col];
    acc = wmma_f32(a, b, acc);
  }
  float bias = l2b[col];
#pragma unroll
  for (int r = 0; r < 8; ++r) {
    int node = n0 + chi + r;
    if (node < N) out[(long long)node * HIDDEN + col] = silu_f(acc[r] + bias);
  }
}

// ---------------------------------------------------------------------------
extern "C" void kernel_launch(void* const* d_in, const int* in_sizes, int n_in,
                              void* d_out, int out_size, void* d_ws,
                              size_t ws_size, hipStream_t stream) {
  const float* x    = (const float*)d_in[0];
  const int*   eidx = (const int*)d_in[1];
  const float* rbf  = (const float*)d_in[2];
  const float* cut  = (const float*)d_in[3];
  const float* w1   = (const float*)d_in[4];
  const float* b1   = (const float*)d_in[5];
  const float* w2   = (const float*)d_in[6];
  const float* b2   = (const float*)d_in[7];
  const float* l1w  = (const float*)d_in[8];
  const float* l1b  = (const float*)d_in[9];
  const float* l2w  = (const float*)d_in[10];
  const float* l2b  = (const float*)d_in[11];
  float* out = (float*)d_out;

  const int E = in_sizes[3];           // cutoff_val is [E]
  const int N = in_sizes[0] / HIDDEN;  // x is [N, HIDDEN]

  float* agg = (float*)d_ws;           // [N, FILTERS] accumulator
  long long aggElems = (long long)N * FILTERS;

  schnet_zero_kernel<<<2048, 256, 0, stream>>>((float4*)agg, aggElems / 4);

  int edgeBlocks = (E + 63) / 64;
  schnet_edge_kernel<<<edgeBlocks, 256, 0, stream>>>(
      x, eidx, rbf, cut, w1, b1, w2, b2, l1w, l1b, agg, E);

  int nodeBlocks = (N + 15) / 16;
  schnet_node_kernel<<<nodeBlocks, 256, 0, stream>>>(agg, l2w, l2b, out, N);
}